// PixelCenterTLoss_77309412138
// MI455X (gfx1250) — compile-verified
//
#include <hip/hip_runtime.h>
#include <hip/hip_bf16.h>

typedef float v2f __attribute__((ext_vector_type(2)));
typedef float v8f __attribute__((ext_vector_type(8)));

#define N_SAMPLES   262144
#define DIMS        128
#define NUM_CLASSES 19
#define CPAD        32            // classes padded to 2 WMMA tiles of 16
#define NB1         256           // pass-1 blocks
#define TPB1        256           // pass-1 threads (8 waves of 32)
#define WAVES1      (TPB1 / 32)
#define SPB1        (N_SAMPLES / NB1)     // 1024 samples/block
#define SPW1        (SPB1 / WAVES1)       // 128 samples/wave
#define NB3         1024
#define TPB3        256

// ---------------------------------------------------------------------------
// Pass 1: class sums via dense one-hot GEMM on the f32 WMMA pipe.
//   sums[c][f] = sum_i onehot[c][i] * X[i][f]
// A = one-hot (16 classes x 4 samples), B = X tile (4 samples x 16 feats),
// accumulate D (16x16 f32) over K in steps of 4. Full f32, bit-exact math.
// ---------------------------------------------------------------------------
__global__ __launch_bounds__(TPB1)
void k1_class_sums(const float* __restrict__ X, const int* __restrict__ T,
                   float* __restrict__ blockSums, float* __restrict__ blockCounts) {
    __shared__ float lsums[CPAD * DIMS];   // 16 KB block-level partial sums
    __shared__ float lcnt[CPAD];

    const int tid = threadIdx.x;
    for (int i = tid; i < CPAD * DIMS; i += TPB1) lsums[i] = 0.0f;
    if (tid < CPAD) lcnt[tid] = 0.0f;
    __syncthreads();

    const int wave = tid >> 5;
    const int lane = tid & 31;
    const int half = lane >> 4;   // selects K pair (A/B) and M half (C/D)
    const int lm   = lane & 15;   // M for A, N for B/C/D

    v8f acc[2][8] = {};           // [class tile][feature tile]

    const int base = blockIdx.x * SPB1 + wave * SPW1;

    for (int g = 0; g < SPW1; g += 4) {
        const int s  = base + g;
        const int t0 = T[s + 2 * half + 0];   // sample for K = 2*half   (vgpr0)
        const int t1 = T[s + 2 * half + 1];   // sample for K = 2*half+1 (vgpr1)

        // exact integer counts in f32 (order-independent, < 2^24)
        if (lm == 0) {
            atomicAdd(&lcnt[t0], 1.0f);
            atomicAdd(&lcnt[t1], 1.0f);
        }

        // A tiles: one-hot columns for the 4 samples of this K-group
        v2f a0, a1;
        a0.x = (t0 == lm)      ? 1.0f : 0.0f;   // classes 0..15
        a0.y = (t1 == lm)      ? 1.0f : 0.0f;
        a1.x = (t0 == lm + 16) ? 1.0f : 0.0f;   // classes 16..31 (19..31 never hit)
        a1.y = (t1 == lm + 16) ? 1.0f : 0.0f;

        const float* row0 = X + (size_t)(s + 2 * half + 0) * DIMS;
        const float* row1 = X + (size_t)(s + 2 * half + 1) * DIMS;

#pragma unroll
        for (int ft = 0; ft < 8; ++ft) {
            v2f b;
            b.x = row0[ft * 16 + lm];   // K = 2*half,   N = ft*16+lm
            b.y = row1[ft * 16 + lm];   // K = 2*half+1, N = ft*16+lm
            acc[0][ft] = __builtin_amdgcn_wmma_f32_16x16x4_f32(
                false, a0, false, b, (short)0, acc[0][ft], false, false);
            acc[1][ft] = __builtin_amdgcn_wmma_f32_16x16x4_f32(
                false, a1, false, b, (short)0, acc[1][ft], false, false);
        }
    }

    // Deterministic intra-block merge: waves accumulate into LDS one at a time.
    // Within a wave all lanes touch distinct (c,f), so plain adds are safe.
    for (int w = 0; w < WAVES1; ++w) {
        if (wave == w) {
#pragma unroll
            for (int ct = 0; ct < 2; ++ct)
#pragma unroll
                for (int ft = 0; ft < 8; ++ft)
#pragma unroll
                    for (int v = 0; v < 8; ++v) {
                        const int c = ct * 16 + half * 8 + v;  // D: M = v + 8*half
                        const int f = ft * 16 + lm;            // D: N = lane&15
                        lsums[c * DIMS + f] += acc[ct][ft][v];
                    }
        }
        __syncthreads();
    }

    float* bs = blockSums + (size_t)blockIdx.x * (CPAD * DIMS);
    for (int i = tid; i < CPAD * DIMS; i += TPB1) bs[i] = lsums[i];
    if (tid < CPAD) blockCounts[blockIdx.x * CPAD + tid] = lcnt[tid];
}

// ---------------------------------------------------------------------------
// Pass 2: reduce block partials in fixed order, divide -> class centers.
// ---------------------------------------------------------------------------
__global__ void k2_centers(const float* __restrict__ blockSums,
                           const float* __restrict__ blockCounts,
                           float* __restrict__ centers) {
    const int idx = blockIdx.x * blockDim.x + threadIdx.x;
    if (idx >= CPAD * DIMS) return;
    const int c = idx / DIMS;
    float s = 0.0f;
    for (int b = 0; b < NB1; ++b) s += blockSums[(size_t)b * (CPAD * DIMS) + idx];
    float cnt = 0.0f;
    for (int b = 0; b < NB1; ++b) cnt += blockCounts[b * CPAD + c];
    centers[idx] = (cnt > 0.0f) ? (s / cnt) : 0.0f;
}

// ---------------------------------------------------------------------------
// Pass 3: per-sample distance to own center (centers staged in LDS),
// fixed-tree block reduction of sqrt distances.
// ---------------------------------------------------------------------------
__global__ __launch_bounds__(TPB3)
void k3_dist(const float* __restrict__ X, const int* __restrict__ T,
             const float* __restrict__ centers, float* __restrict__ blockLoss) {
    __shared__ __align__(16) float lc[NUM_CLASSES * DIMS];   // 9.5 KB
    __shared__ float red[TPB3];
    for (int i = threadIdx.x; i < NUM_CLASSES * DIMS; i += TPB3) lc[i] = centers[i];
    __syncthreads();

    const int i = blockIdx.x * TPB3 + threadIdx.x;   // exact cover of N
    const int t = T[i];
    const float4* xr = (const float4*)(X + (size_t)i * DIMS);
    const float4* cr = (const float4*)(lc + t * DIMS);
    float acc = 0.0f;
#pragma unroll
    for (int j = 0; j < DIMS / 4; ++j) {
        const float4 x = xr[j], c = cr[j];
        const float dx = x.x - c.x, dy = x.y - c.y, dz = x.z - c.z, dw = x.w - c.w;
        acc += dx * dx + dy * dy + dz * dz + dw * dw;
    }
    red[threadIdx.x] = sqrtf(acc);
    __syncthreads();
    for (int sft = TPB3 / 2; sft > 0; sft >>= 1) {
        if (threadIdx.x < sft) red[threadIdx.x] += red[threadIdx.x + sft];
        __syncthreads();
    }
    if (threadIdx.x == 0) blockLoss[blockIdx.x] = red[0];
}

// ---------------------------------------------------------------------------
// Pass 4: final fixed-order reduction and mean.
// ---------------------------------------------------------------------------
__global__ void k4_final(const float* __restrict__ blockLoss, float* __restrict__ out) {
    __shared__ float red[256];
    float s = 0.0f;
    for (int i = threadIdx.x; i < NB3; i += 256) s += blockLoss[i];
    red[threadIdx.x] = s;
    __syncthreads();
    for (int sft = 128; sft > 0; sft >>= 1) {
        if (threadIdx.x < sft) red[threadIdx.x] += red[threadIdx.x + sft];
        __syncthreads();
    }
    if (threadIdx.x == 0) out[0] = red[0] / (float)N_SAMPLES;
}

extern "C" void kernel_launch(void* const* d_in, const int* in_sizes, int n_in,
                              void* d_out, int out_size, void* d_ws, size_t ws_size,
                              hipStream_t stream) {
    (void)in_sizes; (void)n_in; (void)out_size; (void)ws_size;
    const float* X = (const float*)d_in[0];
    const int*   T = (const int*)d_in[1];
    float* ws          = (float*)d_ws;
    float* blockSums   = ws;                                        // NB1*CPAD*DIMS
    float* blockCounts = blockSums + (size_t)NB1 * CPAD * DIMS;     // NB1*CPAD
    float* centers     = blockCounts + NB1 * CPAD;                  // CPAD*DIMS
    float* blockLoss   = centers + CPAD * DIMS;                     // NB3

    k1_class_sums<<<NB1, TPB1, 0, stream>>>(X, T, blockSums, blockCounts);
    k2_centers<<<(CPAD * DIMS + 255) / 256, 256, 0, stream>>>(blockSums, blockCounts, centers);
    k3_dist<<<NB3, TPB3, 0, stream>>>(X, T, centers, blockLoss);
    k4_final<<<1, 256, 0, stream>>>(blockLoss, (float*)d_out);
}